// DeepSeekNeuralMLP_55791625175281
// MI455X (gfx1250) — compile-verified
//
#include <hip/hip_runtime.h>
#include <hip/hip_bf16.h>
#include <math.h>

// ---------------------------------------------------------------------------
// DeepSeekNeuralMLP for MI455X (gfx1250, wave32)
//
//   * Reconstruct gate/up/down weights from spline control points into d_ws.
//   * f32 GEMMs via V_WMMA_F32_16X16X4_F32 (native full-precision matrix op).
//   * Ping-pong LDS double buffering: global->reg prefetch of tile k+1
//     overlaps the WMMA compute of tile k; one barrier per K-step.
//   * Down-GEMM grid ordered N-fastest so each 128-row slice of the 256 MB
//     intermediate stays L2-resident across its 32 re-reads; `inter`/`out`
//     written with non-temporal stores to keep weights resident in L2.
// ---------------------------------------------------------------------------

typedef __attribute__((ext_vector_type(2))) float v2f;
typedef __attribute__((ext_vector_type(8))) float v8f;

#define HID   2048
#define INTER 8192

#define BM 128          // workgroup M tile
#define BN 64           // workgroup N tile
#define BK 16           // K step per LDS stage
#define LDSS 20         // padded LDS row stride in floats (16B-aligned, conflict-free)

// ---------------------------------------------------------------------------
// Kernel 1: Catmull-Rom spline reconstruction of a dense weight vector.
// t-grid computed in double to match the reference's float64 linspace.
// ---------------------------------------------------------------------------
__global__ __launch_bounds__(256)
void nmlp_recon(const float* __restrict__ cp, float* __restrict__ w,
                int n_ctrl, long long n) {
    long long j = (long long)blockIdx.x * blockDim.x + threadIdx.x;
    if (j >= n) return;
    double t = (double)j * (double)(n_ctrl - 1) / (double)(n - 1);
    long long i = (long long)floor(t);
    if (i < 0) i = 0;
    if (i > (long long)n_ctrl - 2) i = (long long)n_ctrl - 2;
    float f = (float)(t - (double)i);
    int i0 = (int)(i - 1 < 0 ? 0 : i - 1);
    int i1 = (int)i;
    int i2 = (int)(i + 1 > (long long)n_ctrl - 1 ? (long long)n_ctrl - 1 : i + 1);
    int i3 = (int)(i + 2 > (long long)n_ctrl - 1 ? (long long)n_ctrl - 1 : i + 2);
    float p0 = cp[i0], p1 = cp[i1], p2 = cp[i2], p3 = cp[i3];
    float f2 = f * f;
    float f3 = f2 * f;
    w[j] = 0.5f * (2.0f * p1 + (p2 - p0) * f
                 + (2.0f * p0 - 5.0f * p1 + 4.0f * p2 - p3) * f2
                 + (3.0f * p1 - p0 - 3.0f * p2 + p3) * f3);
}

// ---------------------------------------------------------------------------
// Kernel 2: fused gate/up GEMM + SiLU*up epilogue (double-buffered).
//   A     : [M, K] row-major (hidden states, K = HID)
//   Wg,Wu : [N, K] row-major (weight, N = INTER)  -> B = W^T
//   inter : [M, N] f32 (non-temporal stores)
// Grid: (M/BM, N/BN), 256 threads = 8 waves; wave grid 4(M) x 2(N),
// each wave owns 32x32 = 2x2 WMMA 16x16 fragments per output matrix.
// ---------------------------------------------------------------------------
__global__ __launch_bounds__(256)
void nmlp_gateup(const float* __restrict__ A,
                 const float* __restrict__ Wg,
                 const float* __restrict__ Wu,
                 float* __restrict__ inter,
                 int M, int N, int K) {
    __shared__ float sA[2][BM * LDSS];
    __shared__ float sG[2][BN * LDSS];
    __shared__ float sU[2][BN * LDSS];

    const int tid   = threadIdx.x;
    const int lane  = tid & 31;
    const int wave  = tid >> 5;     // 0..7
    const int waveM = wave & 3;     // 4 waves across M
    const int waveN = wave >> 2;    // 2 waves across N
    const long long blockM = (long long)blockIdx.x * BM;
    const long long blockN = (long long)blockIdx.y * BN;

    const v8f vzero = {0.f, 0.f, 0.f, 0.f, 0.f, 0.f, 0.f, 0.f};
    v8f accG[2][2], accU[2][2];
#pragma unroll
    for (int ti = 0; ti < 2; ++ti)
#pragma unroll
        for (int tj = 0; tj < 2; ++tj) { accG[ti][tj] = vzero; accU[ti][tj] = vzero; }

    // loader mapping: rows of BK=16 floats, 4 threads per row (float4 each)
    const int ldRow  = tid >> 2;        // 0..63
    const int ldCol4 = (tid & 3) * 4;   // 0,4,8,12

    // fragment addressing (ISA 16x4 f32 A layout; B mirrored via [N][K] LDS)
    const int fr = lane & 15;           // m (or n) within a 16x16 fragment
    const int fk = (lane >> 4) * 2;     // K pair base: 0 or 2

    const float* pA0 = &A [(blockM + ldRow)      * (long long)K + ldCol4];
    const float* pA1 = &A [(blockM + ldRow + 64) * (long long)K + ldCol4];
    const float* pG  = &Wg[(blockN + ldRow)      * (long long)K + ldCol4];
    const float* pU  = &Wu[(blockN + ldRow)      * (long long)K + ldCol4];

    // ---- prologue: stage tile 0 into buffer 0 ----
    {
        float4 a0 = *(const float4*)pA0;
        float4 a1 = *(const float4*)pA1;
        float4 g0 = *(const float4*)pG;
        float4 u0 = *(const float4*)pU;
        *(float4*)&sA[0][ ldRow       * LDSS + ldCol4] = a0;
        *(float4*)&sA[0][(ldRow + 64) * LDSS + ldCol4] = a1;
        *(float4*)&sG[0][ ldRow       * LDSS + ldCol4] = g0;
        *(float4*)&sU[0][ ldRow       * LDSS + ldCol4] = u0;
    }

    int cur = 0;
    for (int k0 = 0; k0 < K; k0 += BK) {
        __syncthreads();   // buffer `cur` staged & previous readers done

        const bool more = (k0 + BK) < K;
        float4 na0, na1, ng, nu;
        if (more) {        // prefetch tile k0+BK into registers (overlaps WMMA)
            na0 = *(const float4*)(pA0 + k0 + BK);
            na1 = *(const float4*)(pA1 + k0 + BK);
            ng  = *(const float4*)(pG  + k0 + BK);
            nu  = *(const float4*)(pU  + k0 + BK);
        }

        const int mBase = waveM * 32;
        const int nBase = waveN * 32;
#pragma unroll
        for (int kk = 0; kk < BK; kk += 4) {
            v2f af[2], gf[2], uf[2];
#pragma unroll
            for (int t = 0; t < 2; ++t) {
                af[t] = *(const v2f*)&sA[cur][(mBase + t * 16 + fr) * LDSS + kk + fk];
                gf[t] = *(const v2f*)&sG[cur][(nBase + t * 16 + fr) * LDSS + kk + fk];
                uf[t] = *(const v2f*)&sU[cur][(nBase + t * 16 + fr) * LDSS + kk + fk];
            }
#pragma unroll
            for (int ti = 0; ti < 2; ++ti)
#pragma unroll
                for (int tj = 0; tj < 2; ++tj) {
                    accG[ti][tj] = __builtin_amdgcn_wmma_f32_16x16x4_f32(
                        false, af[ti], false, gf[tj], (short)0, accG[ti][tj], false, false);
                    accU[ti][tj] = __builtin_amdgcn_wmma_f32_16x16x4_f32(
                        false, af[ti], false, uf[tj], (short)0, accU[ti][tj], false, false);
                }
        }

        if (more) {        // stage prefetched tile into the other buffer
            const int nxt = cur ^ 1;
            *(float4*)&sA[nxt][ ldRow       * LDSS + ldCol4] = na0;
            *(float4*)&sA[nxt][(ldRow + 64) * LDSS + ldCol4] = na1;
            *(float4*)&sG[nxt][ ldRow       * LDSS + ldCol4] = ng;
            *(float4*)&sU[nxt][ ldRow       * LDSS + ldCol4] = nu;
        }
        cur ^= 1;
    }

    // ---- epilogue: silu(gate) * up, non-temporal stores ----
    // C/D layout: vgpr r -> row = r + 8*(lane/16); col = lane%16
    const long long rowBase = blockM + waveM * 32 + ((lane >> 4) * 8);
    const long long colBase = blockN + waveN * 32 + (lane & 15);
#pragma unroll
    for (int ti = 0; ti < 2; ++ti)
#pragma unroll
        for (int tj = 0; tj < 2; ++tj)
#pragma unroll
            for (int r = 0; r < 8; ++r) {
                float g = accG[ti][tj][r];
                float u = accU[ti][tj][r];
                float s = g / (1.0f + __expf(-g));   // silu
                __builtin_nontemporal_store(
                    s * u,
                    &inter[(rowBase + ti * 16 + r) * (long long)N + colBase + tj * 16]);
            }
}

// ---------------------------------------------------------------------------
// Kernel 3: down projection GEMM (double-buffered).
//   A  : [M, K] row-major (inter, K = INTER)
//   Wd : [N, K] row-major (down_w, N = HID)
//   C  : [M, N] f32 output (non-temporal stores)
// Grid: (N/BN, M/BM) -- N varies fastest so the 4 MB A-chunk of a given
// M-block stays L2-resident across all N-blocks, and Wd becomes L2-resident.
// ---------------------------------------------------------------------------
__global__ __launch_bounds__(256)
void nmlp_down(const float* __restrict__ A,
               const float* __restrict__ Wd,
               float* __restrict__ C,
               int M, int N, int K) {
    __shared__ float sA[2][BM * LDSS];
    __shared__ float sB[2][BN * LDSS];

    const int tid   = threadIdx.x;
    const int lane  = tid & 31;
    const int wave  = tid >> 5;
    const int waveM = wave & 3;
    const int waveN = wave >> 2;
    const long long blockM = (long long)blockIdx.y * BM;
    const long long blockN = (long long)blockIdx.x * BN;

    const v8f vzero = {0.f, 0.f, 0.f, 0.f, 0.f, 0.f, 0.f, 0.f};
    v8f acc[2][2];
#pragma unroll
    for (int ti = 0; ti < 2; ++ti)
#pragma unroll
        for (int tj = 0; tj < 2; ++tj) acc[ti][tj] = vzero;

    const int ldRow  = tid >> 2;
    const int ldCol4 = (tid & 3) * 4;
    const int fr = lane & 15;
    const int fk = (lane >> 4) * 2;

    const float* pA0 = &A [(blockM + ldRow)      * (long long)K + ldCol4];
    const float* pA1 = &A [(blockM + ldRow + 64) * (long long)K + ldCol4];
    const float* pB  = &Wd[(blockN + ldRow)      * (long long)K + ldCol4];

    {
        float4 a0 = *(const float4*)pA0;
        float4 a1 = *(const float4*)pA1;
        float4 b0 = *(const float4*)pB;
        *(float4*)&sA[0][ ldRow       * LDSS + ldCol4] = a0;
        *(float4*)&sA[0][(ldRow + 64) * LDSS + ldCol4] = a1;
        *(float4*)&sB[0][ ldRow       * LDSS + ldCol4] = b0;
    }

    int cur = 0;
    for (int k0 = 0; k0 < K; k0 += BK) {
        __syncthreads();

        const bool more = (k0 + BK) < K;
        float4 na0, na1, nb;
        if (more) {
            na0 = *(const float4*)(pA0 + k0 + BK);
            na1 = *(const float4*)(pA1 + k0 + BK);
            nb  = *(const float4*)(pB  + k0 + BK);
        }

        const int mBase = waveM * 32;
        const int nBase = waveN * 32;
#pragma unroll
        for (int kk = 0; kk < BK; kk += 4) {
            v2f af[2], bf[2];
#pragma unroll
            for (int t = 0; t < 2; ++t) {
                af[t] = *(const v2f*)&sA[cur][(mBase + t * 16 + fr) * LDSS + kk + fk];
                bf[t] = *(const v2f*)&sB[cur][(nBase + t * 16 + fr) * LDSS + kk + fk];
            }
#pragma unroll
            for (int ti = 0; ti < 2; ++ti)
#pragma unroll
                for (int tj = 0; tj < 2; ++tj)
                    acc[ti][tj] = __builtin_amdgcn_wmma_f32_16x16x4_f32(
                        false, af[ti], false, bf[tj], (short)0, acc[ti][tj], false, false);
        }

        if (more) {
            const int nxt = cur ^ 1;
            *(float4*)&sA[nxt][ ldRow       * LDSS + ldCol4] = na0;
            *(float4*)&sA[nxt][(ldRow + 64) * LDSS + ldCol4] = na1;
            *(float4*)&sB[nxt][ ldRow       * LDSS + ldCol4] = nb;
        }
        cur ^= 1;
    }

    const long long rowBase = blockM + waveM * 32 + ((lane >> 4) * 8);
    const long long colBase = blockN + waveN * 32 + (lane & 15);
#pragma unroll
    for (int ti = 0; ti < 2; ++ti)
#pragma unroll
        for (int tj = 0; tj < 2; ++tj)
#pragma unroll
            for (int r = 0; r < 8; ++r)
                __builtin_nontemporal_store(
                    acc[ti][tj][r],
                    &C[(rowBase + ti * 16 + r) * (long long)N + colBase + tj * 16]);
}

// ---------------------------------------------------------------------------
// Launch
// ---------------------------------------------------------------------------
extern "C" void kernel_launch(void* const* d_in, const int* in_sizes, int n_in,
                              void* d_out, int out_size, void* d_ws, size_t ws_size,
                              hipStream_t stream) {
    const float* hs      = (const float*)d_in[0];
    const float* gate_cp = (const float*)d_in[1];
    const float* up_cp   = (const float*)d_in[2];
    const float* down_cp = (const float*)d_in[3];
    float* out = (float*)d_out;

    const long long M  = (long long)in_sizes[0] / HID;       // tokens = 8192
    const long long NW = (long long)INTER * HID;             // 16,777,216 per weight

    float* ws    = (float*)d_ws;
    float* Wg    = ws;
    float* Wu    = ws + NW;
    float* Wd    = ws + 2 * NW;
    float* inter = ws + 3 * NW;                              // [M, INTER] f32

    // 1) weight reconstruction (memory-bound one-shot)
    {
        const int threads = 256;
        const int blocks = (int)((NW + threads - 1) / threads);
        nmlp_recon<<<blocks, threads, 0, stream>>>(gate_cp, Wg, in_sizes[1], NW);
        nmlp_recon<<<blocks, threads, 0, stream>>>(up_cp,   Wu, in_sizes[2], NW);
        nmlp_recon<<<blocks, threads, 0, stream>>>(down_cp, Wd, in_sizes[3], NW);
    }

    // 2) fused gate/up GEMM + silu*up (M fastest: hs stays L2-resident)
    {
        dim3 grid((unsigned)(M / BM), (unsigned)(INTER / BN));
        nmlp_gateup<<<grid, 256, 0, stream>>>(hs, Wg, Wu, inter, (int)M, INTER, HID);
    }

    // 3) down projection (N fastest: inter chunk + Wd stay L2-resident)
    {
        dim3 grid((unsigned)(HID / BN), (unsigned)(M / BM));
        nmlp_down<<<grid, 256, 0, stream>>>(inter, Wd, out, (int)M, HID, INTER);
    }
}